// WindowAttention_sparse_74414603370954
// MI455X (gfx1250) — compile-verified
//
#include <hip/hip_runtime.h>
#include <hip/hip_bf16.h>
#include <math.h>
#include <stdint.h>

// ---------------------------------------------------------------------------
// Fused Swin-style window attention for MI455X (gfx1250, wave32, WMMA).
// One workgroup (8 wave32) handles one whole window: QKV GEMM -> biased
// attention (softmax + relu^2 mix) -> attn@V -> output projection, all in
// LDS, using V_WMMA_F32_16X16X4_F32 for every matmul and
// GLOBAL_LOAD_ASYNC_TO_LDS_B128 (ASYNCcnt) for all global->LDS staging.
// ---------------------------------------------------------------------------

typedef __attribute__((ext_vector_type(2))) float v2f;
typedef __attribute__((ext_vector_type(8))) float v8f;

#define NTOK   64
#define CCH    256
#define NHEAD  8
#define HD     32
#define NWIN   2048
#define QSCALE 0.17677669529663689f   // 1/sqrt(32)

// LDS row strides (floats), chosen for bank-conflict-free WMMA frag reads
#define XP 260    // x/q/k/v rows: 260 % 64 = 4 -> lane m hits bank 4m+off
#define WP 80     // weight tile rows: 80 % 64 = 16 -> lane halves 32 banks apart
#define PP 68     // per-head P scratch rows

// float offsets into the single shared array
#define OFF_X 0
#define OFF_Q (64 * XP)          // 16640
#define OFF_K (2 * 64 * XP)      // 33280
#define OFF_V (3 * 64 * XP)      // 49920
#define OFF_W (4 * 64 * XP)      // 66560
#define OFF_R (OFF_W + 64 * WP)  // 71680
#define SMEM_FLOATS (OFF_R + 1800)  // 73480 floats = 293,920 B < 320 KB

// ---- CDNA5 async global->LDS staging (per-lane LDS address in VGPR) -------
__device__ __forceinline__ void async_ld_b128(const float* gptr, float* lptr) {
  // low 32 bits of a generic shared pointer == LDS byte offset
  const unsigned lds = (unsigned)(uintptr_t)lptr;
  asm volatile("global_load_async_to_lds_b128 %0, %1, off"
               :: "v"(lds), "v"(gptr)
               : "memory");
}
__device__ __forceinline__ void wait_async0() {
  asm volatile("s_wait_asynccnt 0" ::: "memory");
}

// ---------------------------------------------------------------------------
// Kernel 0: ego-relative bias.  Needs a global max over all 2048 window
// distances, then gathers ego_table[idx] per (window, head) into d_ws.
// ---------------------------------------------------------------------------
__global__ __launch_bounds__(1024) void ego_bias_kernel(
    const float* __restrict__ aff,        // (2, 2, 3)
    const float* __restrict__ ego_table,  // (16, 8)
    float* __restrict__ ego_ws)           // (2048, 8)
{
  __shared__ float red[2048];
  const int t = threadIdx.x;

  float ex[2], ey[2];
#pragma unroll
  for (int c = 0; c < 2; ++c) {
    const float* A = aff + c * 6;
    ex[c] = A[0] * 128.0f + A[1] * 128.0f + A[2];
    ey[c] = A[3] * 128.0f + A[4] * 128.0f + A[5];
  }

  float rx[2], ry[2], d[2];
#pragma unroll
  for (int i = 0; i < 2; ++i) {
    const int e   = t + i * 1024;
    const int c   = e >> 10;
    const int wdx = e & 1023;
    const int gy  = wdx >> 5, gx = wdx & 31;
    rx[i] = (float)(gx * 8) + 4.0f - ex[c];
    ry[i] = (float)(gy * 8) + 4.0f - ey[c];
    d[i]  = sqrtf(rx[i] * rx[i] + ry[i] * ry[i]);
    red[e] = d[i];
  }
  __syncthreads();
  for (int s = 1024; s > 0; s >>= 1) {
    if (t < s) red[t] = fmaxf(red[t], red[t + s]);
    __syncthreads();
  }
  const float dmax = red[0];

  const float PI = 3.14159265358979323846f;
#pragma unroll
  for (int i = 0; i < 2; ++i) {
    const int e = t + i * 1024;
    int db = (int)(d[i] / (dmax + 1e-6f) * 3.0f);
    float ang = atan2f(ry[i], rx[i]);
    int ab = (int)((ang + PI) / (2.0f * PI) * 3.0f);
    const int idx = db * 4 + ab;
#pragma unroll
    for (int h = 0; h < NHEAD; ++h)
      ego_ws[e * NHEAD + h] = ego_table[idx * NHEAD + h];
  }
}

// ---------------------------------------------------------------------------
// Fused window-attention kernel.  gridDim.x = 2048 windows, 256 threads.
// ---------------------------------------------------------------------------
__global__ __launch_bounds__(256, 1) void winattn_fused(
    const float* __restrict__ x,    // (2048, 64, 256)
    const float* __restrict__ Wq,   // (256, 256)
    const float* __restrict__ bq,   // (256)
    const float* __restrict__ Wkv,  // (256, 512)
    const float* __restrict__ bkv,  // (512)
    const float* __restrict__ Wp,   // (256, 256)
    const float* __restrict__ bp,   // (256)
    const float* __restrict__ rel_table,  // (225, 8)
    const float* __restrict__ wmix,       // (2)
    const float* __restrict__ ego_ws,     // (2048, 8)
    float* __restrict__ out)              // (2048, 64, 256)
{
  __shared__ __align__(16) float smem[SMEM_FLOATS];

  const int b    = blockIdx.x;
  const int tid  = threadIdx.x;
  const int wave = tid >> 5;
  const int lane = tid & 31;
  const int half = lane >> 4;   // WMMA lane half (ISA frag layouts)
  const int l16  = lane & 15;

  float* xs  = smem + OFF_X;
  float* qs  = smem + OFF_Q;
  float* ks  = smem + OFF_K;
  float* vs  = smem + OFF_V;
  float* wls = smem + OFF_W;
  float* rls = smem + OFF_R;

  // ---- phase 0: async-stage x window + rel_table; compute mix weights ----
  const float* xw = x + (size_t)b * NTOK * CCH;
  for (int i = tid; i < NTOK * CCH / 4; i += 256) {   // 16 b128 per thread
    const int row = i >> 6;        // 64 float4 per row
    const int c4  = i & 63;
    async_ld_b128(&xw[row * 64 * 4 + c4 * 4], &xs[row * XP + c4 * 4]);
  }
  for (int i = tid; i < 450; i += 256)                // 1800 floats
    async_ld_b128(&rel_table[i * 4], &rls[i * 4]);

  const float w0 = wmix[0], w1 = wmix[1];
  const float wm = fmaxf(w0, w1);
  const float ew0 = __expf(w0 - wm), ew1 = __expf(w1 - wm);
  const float mix0 = ew0 / (ew0 + ew1);
  const float mix1 = ew1 / (ew0 + ew1);

  // ---- phase 1: QKV projection (x @ [Wq | Wkv]) into q/k/v LDS -----------
  // 8 waves tile each 64x64 output block: wave = (msub, npair)
  const int msub = wave & 3;
  const int m0   = msub * 16;
  const int np   = wave >> 2;
  const int colA = (np * 2 + 0) * 16 + l16;
  const int colB = (np * 2 + 1) * 16 + l16;

  for (int j = 0; j < 12; ++j) {
    const float* Wsrc; const float* bsrc; int ldw, wcol0, dcol0; float* dst;
    if (j < 4) { Wsrc = Wq;  bsrc = bq;  ldw = 256; wcol0 = j * 64;       dst = qs; dcol0 = j * 64; }
    else       { Wsrc = Wkv; bsrc = bkv; ldw = 512; wcol0 = j * 64 - 256;
                 dst = (j < 8) ? ks : vs; dcol0 = (j < 8) ? (j - 4) * 64 : (j - 8) * 64; }

    v8f acc0 = {0, 0, 0, 0, 0, 0, 0, 0};
    v8f acc1 = {0, 0, 0, 0, 0, 0, 0, 0};

    for (int kc = 0; kc < CCH; kc += 64) {
      __syncthreads();   // protect w_lds reuse
      for (int i = tid; i < 64 * 16; i += 256) {   // 4 b128 per thread
        const int row = i >> 4;    // 16 float4 per 64-col row
        const int c4  = i & 15;
        async_ld_b128(&Wsrc[(size_t)(kc + row) * ldw + wcol0 + c4 * 4],
                      &wls[row * WP + c4 * 4]);
      }
      wait_async0();     // also covers the phase-0 x/rel loads on 1st pass
      __syncthreads();

#pragma unroll
      for (int kk = 0; kk < 64; kk += 4) {
        // A frag: x[m][k..k+1] contiguous pair (K = vgpr + 2*half)
        const v2f a = *(const v2f*)&xs[(m0 + l16) * XP + kc + kk + 2 * half];
        v2f bfA, bfB;
        bfA.x = wls[(kk + 2 * half) * WP + colA];
        bfA.y = wls[(kk + 2 * half + 1) * WP + colA];
        acc0 = __builtin_amdgcn_wmma_f32_16x16x4_f32(false, a, false, bfA, (short)0, acc0, false, false);
        bfB.x = wls[(kk + 2 * half) * WP + colB];
        bfB.y = wls[(kk + 2 * half + 1) * WP + colB];
        acc1 = __builtin_amdgcn_wmma_f32_16x16x4_f32(false, a, false, bfB, (short)0, acc1, false, false);
      }
    }

    const float sc = (j < 4) ? QSCALE : 1.0f;   // fold q-scale in
    const float bias0 = bsrc[wcol0 + colA];
    const float bias1 = bsrc[wcol0 + colB];
#pragma unroll
    for (int r = 0; r < 8; ++r) {
      const int row = m0 + r + 8 * half;        // C layout: M = r + 8*half
      dst[row * XP + dcol0 + colA] = (acc0[r] + bias0) * sc;
      dst[row * XP + dcol0 + colB] = (acc1[r] + bias1) * sc;
    }
  }
  __syncthreads();   // q/k/v complete; x now dead (re-used as P scratch)

  // ---- phase 2: attention, one wave per head -----------------------------
  const int h  = wave;
  const int hb = h * HD;
  const float egob = ego_ws[b * NHEAD + h];
  float* ps = smem + OFF_X + h * 16 * PP;   // per-head 16x64 P scratch
  float* ao = ks;                           // attn-out aliases dead k region

  v8f oacc[4][2];
#pragma unroll
  for (int mt = 0; mt < 4; ++mt)
#pragma unroll
    for (int n2 = 0; n2 < 2; ++n2)
      oacc[mt][n2] = (v8f){0, 0, 0, 0, 0, 0, 0, 0};

#pragma unroll
  for (int mt = 0; mt < 4; ++mt) {
    // cached A frags (q rows mt*16..+15, K = 32 channels)
    v2f af[8];
#pragma unroll
    for (int ksi = 0; ksi < 8; ++ksi)
      af[ksi] = *(const v2f*)&qs[(mt * 16 + l16) * XP + hb + ksi * 4 + 2 * half];

    v8f sc4[4];
#pragma unroll
    for (int nt = 0; nt < 4; ++nt) {
      v8f acc = {0, 0, 0, 0, 0, 0, 0, 0};
#pragma unroll
      for (int ksi = 0; ksi < 8; ++ksi) {
        // B = k^T: element [kdim][n] = k[n][kdim] -> contiguous channel pair
        const v2f bf = *(const v2f*)&ks[(nt * 16 + l16) * XP + hb + ksi * 4 + 2 * half];
        acc = __builtin_amdgcn_wmma_f32_16x16x4_f32(false, af[ksi], false, bf, (short)0, acc, false, false);
      }
      sc4[nt] = acc;
    }

    // relative-position + ego bias, per element of the C-layout accumulators
#pragma unroll
    for (int nt = 0; nt < 4; ++nt) {
      const int nk = nt * 16 + l16;
      const int rk = nk >> 3, ck = nk & 7;
#pragma unroll
      for (int r = 0; r < 8; ++r) {
        const int nq = mt * 16 + r + 8 * half;
        const int rq = nq >> 3, cq = nq & 7;
        const int rpi = (rq - rk + 7) * 15 + (cq - ck + 7);
        sc4[nt][r] += rls[rpi * NHEAD + h] + egob;
      }
    }

    // row softmax + relu^2 mix; rows live in 16-lane halves -> xor-butterfly
#pragma unroll
    for (int r = 0; r < 8; ++r) {
      float mx = fmaxf(fmaxf(sc4[0][r], sc4[1][r]), fmaxf(sc4[2][r], sc4[3][r]));
      mx = fmaxf(mx, __shfl_xor(mx, 1, 32));
      mx = fmaxf(mx, __shfl_xor(mx, 2, 32));
      mx = fmaxf(mx, __shfl_xor(mx, 4, 32));
      mx = fmaxf(mx, __shfl_xor(mx, 8, 32));
      float e[4]; float ssum = 0.0f;
#pragma unroll
      for (int nt = 0; nt < 4; ++nt) { e[nt] = __expf(sc4[nt][r] - mx); ssum += e[nt]; }
      ssum += __shfl_xor(ssum, 1, 32);
      ssum += __shfl_xor(ssum, 2, 32);
      ssum += __shfl_xor(ssum, 4, 32);
      ssum += __shfl_xor(ssum, 8, 32);
      const float inv = 1.0f / ssum;
#pragma unroll
      for (int nt = 0; nt < 4; ++nt) {
        const float a0 = e[nt] * inv;
        const float rl = fmaxf(sc4[nt][r], 0.0f);
        sc4[nt][r] = mix0 * a0 + mix1 * rl * rl;
      }
    }

    // stage P tile (C layout -> row-major scratch), wave-private region
#pragma unroll
    for (int nt = 0; nt < 4; ++nt)
#pragma unroll
      for (int r = 0; r < 8; ++r)
        ps[(r + 8 * half) * PP + nt * 16 + l16] = sc4[nt][r];

    // O += P @ V  (K = 64 key tokens)
#pragma unroll
    for (int ksi = 0; ksi < 16; ++ksi) {
      const int k0 = ksi * 4;
      const v2f a = *(const v2f*)&ps[l16 * PP + k0 + 2 * half];
#pragma unroll
      for (int n2 = 0; n2 < 2; ++n2) {
        v2f bf;
        bf.x = vs[(k0 + 2 * half) * XP + hb + n2 * 16 + l16];
        bf.y = vs[(k0 + 2 * half + 1) * XP + hb + n2 * 16 + l16];
        oacc[mt][n2] = __builtin_amdgcn_wmma_f32_16x16x4_f32(false, a, false, bf, (short)0, oacc[mt][n2], false, false);
      }
    }
  }

  // write this head's attn-out columns over the (now dead) k region
#pragma unroll
  for (int mt = 0; mt < 4; ++mt)
#pragma unroll
    for (int n2 = 0; n2 < 2; ++n2)
#pragma unroll
      for (int r = 0; r < 8; ++r)
        ao[(mt * 16 + r + 8 * half) * XP + hb + n2 * 16 + l16] = oacc[mt][n2][r];

  __syncthreads();   // all heads' outputs visible

  // ---- phase 3: output projection (attn_out @ Wp + bp) -> global ---------
  for (int j = 0; j < 4; ++j) {
    const int n0 = j * 64;
    v8f acc0 = {0, 0, 0, 0, 0, 0, 0, 0};
    v8f acc1 = {0, 0, 0, 0, 0, 0, 0, 0};
    for (int kc = 0; kc < CCH; kc += 64) {
      __syncthreads();
      for (int i = tid; i < 64 * 16; i += 256) {
        const int row = i >> 4;
        const int c4  = i & 15;
        async_ld_b128(&Wp[(size_t)(kc + row) * 256 + n0 + c4 * 4],
                      &wls[row * WP + c4 * 4]);
      }
      wait_async0();
      __syncthreads();
#pragma unroll
      for (int kk = 0; kk < 64; kk += 4) {
        const v2f a = *(const v2f*)&ao[(m0 + l16) * XP + kc + kk + 2 * half];
        v2f bfA, bfB;
        bfA.x = wls[(kk + 2 * half) * WP + colA];
        bfA.y = wls[(kk + 2 * half + 1) * WP + colA];
        acc0 = __builtin_amdgcn_wmma_f32_16x16x4_f32(false, a, false, bfA, (short)0, acc0, false, false);
        bfB.x = wls[(kk + 2 * half) * WP + colB];
        bfB.y = wls[(kk + 2 * half + 1) * WP + colB];
        acc1 = __builtin_amdgcn_wmma_f32_16x16x4_f32(false, a, false, bfB, (short)0, acc1, false, false);
      }
    }
    const float b0 = bp[n0 + colA];
    const float b1 = bp[n0 + colB];
#pragma unroll
    for (int r = 0; r < 8; ++r) {
      const int row = m0 + r + 8 * half;
      out[(size_t)(b * NTOK + row) * CCH + n0 + colA] = acc0[r] + b0;
      out[(size_t)(b * NTOK + row) * CCH + n0 + colB] = acc1[r] + b1;
    }
  }
}

// ---------------------------------------------------------------------------
extern "C" void kernel_launch(void* const* d_in, const int* in_sizes, int n_in,
                              void* d_out, int out_size, void* d_ws, size_t ws_size,
                              hipStream_t stream) {
  (void)in_sizes; (void)n_in; (void)out_size; (void)ws_size;
  const float* x    = (const float*)d_in[0];
  const float* aff  = (const float*)d_in[1];
  const float* Wq   = (const float*)d_in[2];
  const float* bq   = (const float*)d_in[3];
  const float* Wkv  = (const float*)d_in[4];
  const float* bkv  = (const float*)d_in[5];
  const float* Wp   = (const float*)d_in[6];
  const float* bp   = (const float*)d_in[7];
  const float* rel  = (const float*)d_in[8];
  const float* ego  = (const float*)d_in[9];
  const float* wmix = (const float*)d_in[10];
  float* out    = (float*)d_out;
  float* ego_ws = (float*)d_ws;   // 2048*8 floats

  ego_bias_kernel<<<dim3(1), dim3(1024), 0, stream>>>(aff, ego, ego_ws);
  winattn_fused<<<dim3(NWIN), dim3(256), 0, stream>>>(
      x, Wq, bq, Wkv, bkv, Wp, bp, rel, wmix, ego_ws, out);
}